// OptNet_76579266888161
// MI455X (gfx1250) — compile-verified
//
#include <hip/hip_runtime.h>
#include <hip/hip_bf16.h>

// ---------------------------------------------------------------------------
// OptNet forward pass for MI455X (gfx1250, wave32).
//   GEMMs:  V_WMMA_F32_16X16X4_F32. Each 256-thread block computes a
//           128(M) x 64(N) tile. The weight slab W[0:K, col0:col0+64] is
//           staged ONCE into LDS -- pre-swizzled into B-fragment layout --
//           using global_load_async_to_lds_b32 (ASYNCcnt path). Inner loop:
//           conflict-free ds_load_b64 for B + register-double-buffered
//           global_load_b64 for A, 16 WMMAs per prefetch chunk.
//   QP:     per-batch 64x64 KKT LU (partial pivoting) + 10 refinement steps,
//           one 64-thread workgroup per batch element, everything in LDS.
// ---------------------------------------------------------------------------

typedef float v8f __attribute__((ext_vector_type(8)));
typedef float v2f __attribute__((ext_vector_type(2)));

#define REG  3.0f
#define QSZ  32
#define KSZ  64          // 2*QSZ
#define KP   65          // LDS row pitch (pad to dodge bank conflicts)
#define REFINE_ITERS 10

// ---------------------------------------------------------------------------
// Fused GEMM:  C[M,N] = act( A[M,K] @ W[K,N] + bias[N] (+ res[M,N]) )
//   KDIM : compile-time K (32/128/256) -> immediate-offset A addressing
//   NT   : 16-wide N tiles per wave (4 -> 64-col block tile, 2 -> 32-col)
// Fragment layouts (wave32):
//   A (16x4 f32, 2 VGPRs): lane = m + 16*(k>=2), vgpr = k&1
//   B (4x16  f32, 2 VGPRs): lane = n + 16*(k>=2), vgpr = k&1
//   C/D (16x16 f32, 8 VGPRs): lane n = L&15, row m = vgpr + 8*(L>>4)
// LDS B slab is stored pre-swizzled: fragment (tile i, k-step s) for lane L
// lives at dwords ((i*STEPS+s)*32 + L)*2 .. +1  -> ds_load_b64, 0 conflicts.
// ---------------------------------------------------------------------------
template <int KDIM, int NT>
__global__ __launch_bounds__(256) void gemm_wmma_f32(
    const float* __restrict__ A, const float* __restrict__ W,
    const float* __restrict__ bias, const float* __restrict__ res,
    float* __restrict__ C, int M, int N, int act_swish)
{
    constexpr int STEPS = KDIM / 4;          // k-steps of 4
    constexpr int CH    = 4;                 // k-steps per prefetch chunk
    static_assert(STEPS % CH == 0, "K must be a multiple of 16");
    constexpr int SLAB  = STEPS * NT * 64;   // dwords in LDS B slab

    __shared__ float bsm[SLAB];

    const int tid  = threadIdx.x;
    const int lane = tid & 31;
    const int wv   = tid >> 5;                       // 0..7
    const int row0 = (blockIdx.x << 7) + (wv << 4);  // this wave's 16 M-rows
    const int col0 = blockIdx.y * (16 * NT);         // block's 64/32 N-cols

    // ---- stage swizzled W slab into LDS (async DMA, per-lane gather) ----
    for (int e = tid; e < SLAB; e += 256) {
        const int i    = e / (STEPS * 64);           // N tile
        const int r    = e - i * (STEPS * 64);
        const int s    = r >> 6;                     // k-step
        const int q    = r & 63;
        const int ln   = q >> 1;                     // target lane
        const int half = q & 1;                      // vgpr half (k parity)
        const int k    = 4 * s + ((ln >> 4) << 1) + half;
        const int col  = col0 + (i << 4) + (ln & 15);
        const float* gp = W + (size_t)k * N + col;
        const unsigned lo = (unsigned)(uintptr_t)(&bsm[e]);   // LDS byte offset
        asm volatile("global_load_async_to_lds_b32 %0, %1, off"
                     :: "v"(lo), "v"((unsigned long long)(uintptr_t)gp)
                     : "memory");
    }
    asm volatile("s_wait_asynccnt 0x0" ::: "memory");
    __syncthreads();

    // ---- main loop: A double-buffered from global, B from LDS ----
    const int m  = lane & 15;
    const int kh = (lane >> 4) << 1;                 // 0 or 2

    v8f acc[NT];
#pragma unroll
    for (int i = 0; i < NT; ++i) acc[i] = (v8f){};

    const float* __restrict__ arow = A + (size_t)(row0 + m) * KDIM + kh;

    v2f af[CH], afn[CH];
#pragma unroll
    for (int j = 0; j < CH; ++j) {
        af[j].x = arow[4 * j];
        af[j].y = arow[4 * j + 1];
    }

    for (int c = 0; c < STEPS / CH; ++c) {
        const int sbase = c * CH;
        if (c + 1 < STEPS / CH) {                    // prefetch next chunk
#pragma unroll
            for (int j = 0; j < CH; ++j) {
                const int k0 = (sbase + CH + j) * 4;
                afn[j].x = arow[k0];
                afn[j].y = arow[k0 + 1];
            }
        }
#pragma unroll
        for (int j = 0; j < CH; ++j) {
#pragma unroll
            for (int i = 0; i < NT; ++i) {
                const v2f bf = *(const v2f*)
                    &bsm[(((i * STEPS) + sbase + j) * 32 + lane) * 2];
                acc[i] = __builtin_amdgcn_wmma_f32_16x16x4_f32(
                    false, af[j], false, bf, (short)0, acc[i], false, false);
            }
        }
#pragma unroll
        for (int j = 0; j < CH; ++j) af[j] = afn[j];
    }

    // ---- epilogue: bias (+residual) (+swish) ----
    const int n    = lane & 15;
    const int mrow = (lane >> 4) << 3;               // 0 or 8
#pragma unroll
    for (int i = 0; i < NT; ++i) {
        const int col = col0 + (i << 4) + n;
        const float bv = bias[col];
#pragma unroll
        for (int v = 0; v < 8; ++v) {
            const size_t idx = (size_t)(row0 + mrow + v) * N + col;
            float xv = acc[i][v] + bv;
            if (res) xv += res[idx];
            if (act_swish) xv = xv / (1.0f + __expf(-xv));
            C[idx] = xv;
        }
    }
}

// ---------------------------------------------------------------------------
// Batched equality-constrained QP via regularized KKT LU + iterative refine.
// One workgroup (64 threads = 2 waves) per batch element.
//   K    = [[Q, A^T],[A, 0]]            (unregularized, for refinement)
//   Kreg = K + REG*diag(+1...,-1...) -> LU with partial pivoting (in LDS)
//   s = lu_solve(rhs); 10x { r = rhs - K s; s += lu_solve(r); }
//   out = s[0:32]
// ---------------------------------------------------------------------------
__device__ inline void lu_solve64(float (*LU)[KP], const int* piv, float* x, int t)
{
    if (t == 0) {                       // LAPACK laswp (serial, thread 0)
        for (int k = 0; k < KSZ; ++k) {
            const int p = piv[k];
            const float tmp = x[k]; x[k] = x[p]; x[p] = tmp;
        }
    }
    __syncthreads();
    for (int k = 0; k < KSZ; ++k) {     // forward, unit lower
        const float xk = x[k];
        if (t > k) x[t] -= LU[t][k] * xk;
        __syncthreads();
    }
    for (int k = KSZ - 1; k >= 0; --k) {// backward, upper
        if (t == k) x[k] /= LU[k][k];
        __syncthreads();
        const float xk = x[k];
        if (t < k) x[t] -= LU[t][k] * xk;
        __syncthreads();
    }
}

__global__ __launch_bounds__(64) void qp_solve_kernel(
    const float* __restrict__ Abatch, const float* __restrict__ bbatch,
    const float* __restrict__ Q, const float* __restrict__ c,
    float* __restrict__ sout)
{
    __shared__ float Kmat[KSZ][KP];   // unregularized K (refinement matvec)
    __shared__ float LU[KSZ][KP];     // in-place LU of Kreg
    __shared__ float rhs[KSZ];
    __shared__ float s[KSZ];
    __shared__ float y[KSZ];
    __shared__ float redv[KSZ];
    __shared__ int   redi[KSZ];
    __shared__ int   piv[KSZ];

    const int t = threadIdx.x;                 // 0..63, owns row t
    const int b = blockIdx.x;
    const float* __restrict__ Ab = Abatch + (size_t)b * QSZ * QSZ;

    for (int j = 0; j < KSZ; ++j) {            // build K and Kreg row t
        float v;
        if (t < QSZ) v = (j < QSZ) ? Q[t * QSZ + j] : Ab[(j - QSZ) * QSZ + t];
        else         v = (j < QSZ) ? Ab[(t - QSZ) * QSZ + j] : 0.0f;
        Kmat[t][j] = v;
        if (j == t) v += (t < QSZ) ? REG : -REG;
        LU[t][j] = v;
    }
    rhs[t] = (t < QSZ) ? -c[t] : bbatch[(size_t)b * QSZ + (t - QSZ)];
    __syncthreads();

    // ---- LU factorization with partial pivoting ----
    for (int k = 0; k < KSZ; ++k) {
        redv[t] = (t >= k) ? fabsf(LU[t][k]) : -1.0f;
        redi[t] = t;
        __syncthreads();
#pragma unroll
        for (int off = 32; off > 0; off >>= 1) {
            if (t < off && redv[t + off] > redv[t]) {
                redv[t] = redv[t + off];
                redi[t] = redi[t + off];
            }
            __syncthreads();
        }
        const int p = redi[0];
        if (t == 0) piv[k] = p;
        if (p != k) {                          // swap rows k,p; thread t = col t
            const float tmp = LU[k][t]; LU[k][t] = LU[p][t]; LU[p][t] = tmp;
        }
        __syncthreads();
        const float pv = LU[k][k];
        if (t > k) {
            const float l = LU[t][k] / pv;
            LU[t][k] = l;
            for (int j = k + 1; j < KSZ; ++j) LU[t][j] -= l * LU[k][j];
        }
        __syncthreads();
    }

    // ---- initial solve ----
    y[t] = rhs[t];
    __syncthreads();
    lu_solve64(LU, piv, y, t);
    s[t] = y[t];
    __syncthreads();

    // ---- iterative refinement against unregularized K ----
    for (int it = 0; it < REFINE_ITERS; ++it) {
        float acc = 0.0f;
        for (int j = 0; j < KSZ; ++j) acc += Kmat[t][j] * s[j];
        y[t] = rhs[t] - acc;
        __syncthreads();
        lu_solve64(LU, piv, y, t);
        s[t] += y[t];
        __syncthreads();
    }

    if (t < QSZ) sout[(size_t)b * QSZ + t] = s[t];
}

// ---------------------------------------------------------------------------
// Host side
// ---------------------------------------------------------------------------
template <int KDIM, int NT>
static inline void launch_gemm(const float* A, const float* W, const float* bias,
                               const float* res, float* C, int M, int N,
                               int act, hipStream_t stream)
{
    dim3 grid(M / 128, N / (16 * NT));
    gemm_wmma_f32<KDIM, NT><<<grid, 256, 0, stream>>>(A, W, bias, res, C, M, N, act);
}

extern "C" void kernel_launch(void* const* d_in, const int* in_sizes, int n_in,
                              void* d_out, int out_size, void* d_ws, size_t ws_size,
                              hipStream_t stream)
{
    const float* x     = (const float*)d_in[0];
    const float* W1    = (const float*)d_in[1];
    const float* b1    = (const float*)d_in[2];
    const float* W2    = (const float*)d_in[3];
    const float* b2    = (const float*)d_in[4];
    const float* Wa1   = (const float*)d_in[5];
    const float* ba1   = (const float*)d_in[6];
    const float* Wb1   = (const float*)d_in[7];
    const float* bb1   = (const float*)d_in[8];
    const float* qmat1 = (const float*)d_in[9];
    const float* cvec1 = (const float*)d_in[10];
    const float* W3    = (const float*)d_in[11];
    const float* b3    = (const float*)d_in[12];
    const float* W4    = (const float*)d_in[13];
    const float* b4    = (const float*)d_in[14];
    const float* Wb2   = (const float*)d_in[15];
    const float* bb2   = (const float*)d_in[16];
    const float* qmat2 = (const float*)d_in[17];
    const float* cvec2 = (const float*)d_in[18];
    const float* W5    = (const float*)d_in[19];
    const float* b5    = (const float*)d_in[20];
    const float* W6    = (const float*)d_in[21];
    const float* b6    = (const float*)d_in[22];
    float* out = (float*)d_out;

    const int OBS = 256;
    const int M = in_sizes[0] / OBS;          // batch = 16384

    // workspace layout (floats), buffers reused across lifetimes:
    //   R0: y1 / y3 / y5   (M*256) | R1: y2 / y4 (M*128)
    //   R2: A1 / A2 (M*1024)       | R3: bv (M*32) | R4: s (M*32)
    float* R0 = (float*)d_ws;
    float* R1 = R0 + (size_t)M * 256;
    float* R2 = R1 + (size_t)M * 128;
    float* R3 = R2 + (size_t)M * 1024;
    float* R4 = R3 + (size_t)M * 32;
    (void)ws_size; (void)n_in; (void)out_size;

    // y1 = swish(x @ W1 + b1)                      [M,256]
    launch_gemm<256, 4>(x,  W1, b1, nullptr, R0, M, 256, 1, stream);
    // y2 = swish(y1 @ W2 + b2)                     [M,128]
    launch_gemm<256, 4>(R0, W2, b2, nullptr, R1, M, 128, 1, stream);
    // A1 = y2 @ Wa1 + ba1                          [M,1024]
    launch_gemm<128, 4>(R1, Wa1, ba1, nullptr, R2, M, 1024, 0, stream);
    // bv1 = y2 @ Wb1 + bb1                         [M,32]
    launch_gemm<128, 2>(R1, Wb1, bb1, nullptr, R3, M, 32, 0, stream);
    // s1 = qp_solve(A1, bv1, Q1, c1)               [M,32]
    qp_solve_kernel<<<M, 64, 0, stream>>>(R2, R3, qmat1, cvec1, R4);
    // y3 = swish(s1 @ W3 + b3 + y2)                [M,128] -> R0
    launch_gemm<32, 4>(R4, W3, b3, R1, R0, M, 128, 1, stream);
    // y4 = swish(y3 @ W4 + b4)                     [M,128] -> R1
    launch_gemm<128, 4>(R0, W4, b4, nullptr, R1, M, 128, 1, stream);
    // A2 = y4 @ Wa1 + ba1 (reference reuses Wa1)   [M,1024]
    launch_gemm<128, 4>(R1, Wa1, ba1, nullptr, R2, M, 1024, 0, stream);
    // bv2 = y4 @ Wb2 + bb2                         [M,32]
    launch_gemm<128, 2>(R1, Wb2, bb2, nullptr, R3, M, 32, 0, stream);
    // s2 = qp_solve(A2, bv2, Q2, c2)               [M,32]
    qp_solve_kernel<<<M, 64, 0, stream>>>(R2, R3, qmat2, cvec2, R4);
    // y5 = swish(s2 @ W5 + b5 + y4)                [M,128] -> R0
    launch_gemm<32, 4>(R4, W5, b5, R1, R0, M, 128, 1, stream);
    // out = y5 @ W6 + b6                           [M,32]
    launch_gemm<128, 2>(R0, W6, b6, nullptr, out, M, 32, 0, stream);
}